// MaxYager2d_52802327937310
// MI455X (gfx1250) — compile-verified
//
#include <hip/hip_runtime.h>
#include <math.h>

#define B_      4
#define C_      32
#define H_      66
#define W_      66
#define S_      64
#define F_      32
#define KK      3
#define TILE    16
#define XT      18              // TILE + KK - 1
#define XTE     (XT * XT)       // 324
#define CCH     8               // channels per LDS chunk
#define NCHUNK  (C_ / CCH)      // 4
#define THREADS 256

#if defined(__has_builtin)
#  if __has_builtin(__builtin_amdgcn_global_load_async_to_lds_b32)
#    define HAVE_ASYNC_B32 1
#  endif
#  if __has_builtin(__builtin_amdgcn_s_wait_asynccnt)
#    define HAVE_WAIT_ASYNC 1
#  endif
#  if __has_builtin(__builtin_amdgcn_logf) && __has_builtin(__builtin_amdgcn_exp2f)
#    define HAVE_HW_TRANS 1
#  endif
#endif

typedef __attribute__((address_space(1))) int gas_int;   // global dword
typedef __attribute__((address_space(3))) int las_int;   // LDS dword

// Async DMA of one dword global -> LDS (gfx1250 GLOBAL_LOAD_ASYNC_TO_LDS_B32,
// tracked with ASYNCcnt). Pointer casts go through integers: generic LDS
// addresses carry the LDS offset in the low 32 bits (ISA 10.2 aperture rules),
// and generic global addresses are identity-mapped.
__device__ __forceinline__ void async_copy_b32(const float* g, float* l) {
#if defined(HAVE_ASYNC_B32)
    __builtin_amdgcn_global_load_async_to_lds_b32(
        (gas_int*)(unsigned long long)g,
        (las_int*)(unsigned int)(unsigned long long)l,
        0, 0);
#else
    *l = *g;  // fallback: plain vmem load + ds store
#endif
}

__device__ __forceinline__ void wait_async_all() {
#if defined(HAVE_ASYNC_B32)
#  if defined(HAVE_WAIT_ASYNC)
    __builtin_amdgcn_s_wait_asynccnt(0);
#  else
    asm volatile("s_wait_asynccnt 0" ::: "memory");
#  endif
#endif
}

// m^(2/3) for strictly positive finite m via hardware v_log_f32 / v_exp_f32.
__device__ __forceinline__ float pow_two_thirds(float m) {
#if defined(HAVE_HW_TRANS)
    return __builtin_amdgcn_exp2f(0.66666667f * __builtin_amdgcn_logf(m));
#else
    return exp2f(0.66666667f * log2f(m));
#endif
}

// out[b,f,y,x] = max(1 - (min_j (a[j,(y,x)] + bw[j,f]))^(2/3), 0)
// a  = (1-x)^1.5 over 3x3 unfold windows, bw = (1-w)^1.5.
__global__ __launch_bounds__(THREADS, 2)
void yager_tropical_gemm(const float* __restrict__ xg,
                         const float* __restrict__ wg,
                         float* __restrict__ out) {
    __shared__ float s_x[CCH * XTE];        // 2592 floats: 8ch x 18x18 tile
    __shared__ float s_bw[CCH * 9 * F_];    // 2304 floats: 72 j-rows x 32 f

    const int tid = threadIdx.x;
    const int bb  = blockIdx.x >> 4;        // batch
    const int tl  = blockIdx.x & 15;        // 4x4 tile grid over 64x64
    const int ty0 = (tl >> 2) << 4;
    const int tx0 = (tl & 3) << 4;
    const int txr = tid & 15;
    const int tyr = tid >> 4;

    float acc[F_];
#pragma unroll
    for (int f = 0; f < F_; ++f) acc[f] = 3.0e38f;

#pragma unroll 1
    for (int ch = 0; ch < NCHUNK; ++ch) {
        const int c0 = ch * CCH;
        __syncthreads();  // previous chunk's compute done before overwrite

        // ---- stage input tile: async global -> LDS --------------------
#pragma unroll 1
        for (int e = tid; e < CCH * XTE; e += THREADS) {
            const int cl  = e / XTE;
            const int rem = e - cl * XTE;
            const int r   = rem / XT;
            const int col = rem - r * XT;
            const float* gp = xg +
                ((((size_t)bb * C_ + (c0 + cl)) * H_ + (ty0 + r)) * W_ +
                 (tx0 + col));
            async_copy_b32(gp, &s_x[e]);
        }

        // ---- stage weights: load + transform (1-w)^1.5 ----------------
        const float* wrow = wg + (size_t)c0 * 9 * F_;  // j = c*9+kh*3+kw rows
#pragma unroll 1
        for (int e = tid; e < CCH * 9 * F_; e += THREADS) {
            const float t = 1.0f - wrow[e];
            s_bw[e] = t * __builtin_sqrtf(t);
        }

        wait_async_all();   // this wave's async LDS writes complete
        __syncthreads();    // + dscnt wait: all waves' staging visible

        // ---- transform x tile in place: a = (1-x)^1.5 -----------------
#pragma unroll 1
        for (int e = tid; e < CCH * XTE; e += THREADS) {
            const float t = 1.0f - s_x[e];
            s_x[e] = t * __builtin_sqrtf(t);
        }
        __syncthreads();

        // ---- tropical (min,+) accumulation ----------------------------
        // Per j: 1 per-lane LDS read + 8 broadcast ds_load_b128 + 64 VALU.
#pragma unroll 1
        for (int cl = 0; cl < CCH; ++cl) {
            const float* xs = &s_x[cl * XTE + tyr * XT + txr];
            const float* ws = &s_bw[cl * 9 * F_];
#pragma unroll
            for (int kh = 0; kh < KK; ++kh) {
#pragma unroll
                for (int kw = 0; kw < KK; ++kw) {
                    const float av = xs[kh * XT + kw];                 // per-lane
                    const float4* wr =
                        (const float4*)&ws[(kh * KK + kw) * F_];       // broadcast
#pragma unroll
                    for (int q = 0; q < F_ / 4; ++q) {
                        const float4 w4 = wr[q];
                        acc[q * 4 + 0] = fminf(acc[q * 4 + 0], av + w4.x);
                        acc[q * 4 + 1] = fminf(acc[q * 4 + 1], av + w4.y);
                        acc[q * 4 + 2] = fminf(acc[q * 4 + 2], av + w4.z);
                        acc[q * 4 + 3] = fminf(acc[q * 4 + 3], av + w4.w);
                    }
                }
            }
        }
    }

    // ---- one transcendental per output: max(1 - m^(2/3), 0) -----------
    // m in (0, 2] strictly positive -> raw v_log_f32 / v_exp_f32 are safe.
    const int y = ty0 + tyr;
    const int xcol = tx0 + txr;
    float* op = out + (((size_t)bb * F_) * S_ + y) * S_ + xcol;
#pragma unroll
    for (int f = 0; f < F_; ++f) {
        const float m = acc[f];
        const float r = 1.0f - pow_two_thirds(m);
        op[(size_t)f * (S_ * S_)] = fmaxf(r, 0.0f);
    }
}

extern "C" void kernel_launch(void* const* d_in, const int* in_sizes, int n_in,
                              void* d_out, int out_size, void* d_ws, size_t ws_size,
                              hipStream_t stream) {
    (void)in_sizes; (void)n_in; (void)d_ws; (void)ws_size; (void)out_size;
    const float* x = (const float*)d_in[0];   // [4,32,66,66] f32
    const float* w = (const float*)d_in[1];   // [288,32] f32
    float* out = (float*)d_out;               // [4,32,64,64] f32

    dim3 grid(B_ * 16);       // 4 batches x 16 output tiles of 16x16
    dim3 block(THREADS);      // 8 wave32s
    hipLaunchKernelGGL(yager_tropical_gemm, grid, block, 0, stream, x, w, out);
}